// MeanPooling_47047071760692
// MI455X (gfx1250) — compile-verified
//
#include <hip/hip_runtime.h>

// GCN mean-pool: out[r,:] = sum_{e: rows[e]==r} x[cols[e],:] / max(deg[r],1)
// N=100000 nodes, E=3200000 edges, D=64 features. Scatter-add workload:
// L2-resident (51 MB total state << 192 MB L2), limited by L2 atomic
// throughput, not FLOPs -> no WMMA; CDNA5 features used: wave32 per-edge
// mapping + async global->LDS staging of the edge-index streams (ASYNCcnt).

#define D_FEAT 64
#define TPB    256   // threads per block (8 wave32 waves)
#define EPB    256   // edges staged per block

#if defined(__has_builtin)
#  if __has_builtin(__builtin_amdgcn_global_load_async_to_lds_b32) && \
      __has_builtin(__builtin_amdgcn_s_wait_asynccnt)
#    define USE_ASYNC_LDS 1
#  endif
#endif
#ifndef USE_ASYNC_LDS
#  define USE_ASYNC_LDS 0
#endif

// ---------------------------------------------------------------------------
// Kernel: edge-parallel scatter-add + degree count.  (First in file so the
// disasm snippet shows the async-LDS staging and atomic lowering.)
// Block stages EPB edges of (rows, cols) into LDS via async global->LDS
// copies, then each wave32 processes 32 edges: lane l handles features
// {2l, 2l+1} as one b64 gather + two f32 atomic adds; lane 0 counts degree.
// ---------------------------------------------------------------------------
__global__ void gcn_scatter_kernel(const float* __restrict__ x,
                                   const int* __restrict__ rows,
                                   const int* __restrict__ cols,
                                   float* __restrict__ out,
                                   float* __restrict__ deg,
                                   int n_edges) {
  __shared__ int sRow[EPB];
  __shared__ int sCol[EPB];

  const int tid  = threadIdx.x;
  const int base = blockIdx.x * EPB;
  const int e0   = base + tid;

#if USE_ASYNC_LDS
  if (e0 < n_edges) {
    __builtin_amdgcn_global_load_async_to_lds_b32(
        (int*)(rows + e0), &sRow[tid], 0, 0);
    __builtin_amdgcn_global_load_async_to_lds_b32(
        (int*)(cols + e0), &sCol[tid], 0, 0);
  }
  __builtin_amdgcn_s_wait_asynccnt(0);
#else
  if (e0 < n_edges) {
    sRow[tid] = rows[e0];
    sCol[tid] = cols[e0];
  }
#endif
  __syncthreads();

  const int waveId = tid >> 5;   // 0..7
  const int lane   = tid & 31;   // 0..31 (wave32)
  const int eBase  = base + waveId * 32;
  const int sBase  = waveId * 32;

  // Computed trip count: no per-iteration bounds compare in the hot loop.
  int rem = n_edges - eBase;
  if (rem <= 0) return;
  const int nIter = rem < 32 ? rem : 32;

  for (int j = 0; j < nIter; ++j) {
    const int r = sRow[sBase + j];   // uniform across wave -> broadcast
    const int c = sCol[sBase + j];

    const float2 v = *reinterpret_cast<const float2*>(
        x + (size_t)c * D_FEAT + lane * 2);
    float* dst = out + (size_t)r * D_FEAT + lane * 2;
    atomicAdd(dst + 0, v.x);        // global_atomic_add_f32, resolves in L2
    atomicAdd(dst + 1, v.y);
    if (lane == 0) atomicAdd(deg + r, 1.0f);
  }
}

// ---------------------------------------------------------------------------
// Kernel: zero the accumulator (d_out) and the degree array (d_ws).
// ---------------------------------------------------------------------------
__global__ void gcn_zero_kernel(float4* __restrict__ out4, int n4,
                                float* __restrict__ deg, int n_nodes) {
  int i = blockIdx.x * blockDim.x + threadIdx.x;
  int stride = gridDim.x * blockDim.x;
  float4 z = make_float4(0.f, 0.f, 0.f, 0.f);
  for (int j = i; j < n4; j += stride) out4[j] = z;
  for (int j = i; j < n_nodes; j += stride) deg[j] = 0.f;
}

// ---------------------------------------------------------------------------
// Kernel: normalize. 16 threads per node, float4 per thread.
// ---------------------------------------------------------------------------
__global__ void gcn_normalize_kernel(float* __restrict__ out,
                                     const float* __restrict__ deg,
                                     int n_nodes) {
  const int t = blockIdx.x * blockDim.x + threadIdx.x;
  const int n = t >> 4;            // node
  if (n >= n_nodes) return;
  const int f = (t & 15) * 4;      // feature quad

  const float d   = deg[n];
  const float inv = 1.0f / (d == 0.f ? 1.f : d);

  float4* p = reinterpret_cast<float4*>(out + (size_t)n * D_FEAT + f);
  float4 v = *p;
  v.x *= inv; v.y *= inv; v.z *= inv; v.w *= inv;
  *p = v;
}

// ---------------------------------------------------------------------------
extern "C" void kernel_launch(void* const* d_in, const int* in_sizes, int n_in,
                              void* d_out, int out_size, void* d_ws, size_t ws_size,
                              hipStream_t stream) {
  const float* x    = (const float*)d_in[0];   // [N, 64] f32
  const int*   rows = (const int*)d_in[1];     // [E] i32
  const int*   cols = (const int*)d_in[2];     // [E] i32
  float*       out  = (float*)d_out;           // [N, 64] f32
  float*       deg  = (float*)d_ws;            // [N] f32 scratch

  const int n_nodes = in_sizes[0] / D_FEAT;
  const int n_edges = in_sizes[1];
  const int n4      = out_size / 4;

  // 1) zero accumulator + degrees (d_out/d_ws are poisoned by the harness)
  gcn_zero_kernel<<<1024, TPB, 0, stream>>>((float4*)out, n4, deg, n_nodes);

  // 2) scatter-add + degree count
  const int nblk = (n_edges + EPB - 1) / EPB;   // 12500 blocks for E=3.2M
  gcn_scatter_kernel<<<nblk, TPB, 0, stream>>>(x, rows, cols, out, deg, n_edges);

  // 3) normalize by degree (deg==0 -> 1)
  const int nthr = n_nodes * (D_FEAT / 4);
  gcn_normalize_kernel<<<(nthr + TPB - 1) / TPB, TPB, 0, stream>>>(out, deg, n_nodes);
}